// SparseGATConv_82154134438199
// MI455X (gfx1250) — compile-verified
//
#include <hip/hip_runtime.h>
#include <hip/hip_bf16.h>

// ---------------------------------------------------------------------------
// Sparse GAT forward for MI455X (gfx1250, wave32).
//   h   = feat @ W               (WMMA f32 16x16x4, W transposed+padded in LDS)
//   el  = sum(h*attn_l), er = sum(h*attn_r)
//   s   = leakyrelu(el[src]+er[trg]);  s -= global_max(s)
//   e   = exp(s); denom = segsum(e, trg); attn = e/(denom[trg]+eps)
//   out = segsum(h[src]*attn, trg)
// ---------------------------------------------------------------------------

typedef __attribute__((ext_vector_type(2))) float v2f;
typedef __attribute__((ext_vector_type(8))) float v8f;

#define GAT_HEADS 4
#define GAT_OUTF  16
#define GAT_INF   128
#define GAT_HF    64          // HEADS*OUT_FEATS
#define GAT_NEG   0.2f
#define GAT_EPS   1e-16f
#define WT_STRIDE 132         // padded K-stride for transposed W in LDS
                              // bank = (4*col + 2*halfk) % 64 -> conflict-free b64 reads

// ---------------- ordered-uint encoding for float atomic max ---------------
__device__ __forceinline__ unsigned f32_to_ordered(float f) {
    unsigned u = __float_as_uint(f);
    return (u & 0x80000000u) ? ~u : (u | 0x80000000u);
}
__device__ __forceinline__ float ordered_to_f32(unsigned u) {
    unsigned b = (u & 0x80000000u) ? (u & 0x7fffffffu) : ~u;
    return __uint_as_float(b);
}

// ---------------------------- K0: init -------------------------------------
__global__ void gat_init(float* __restrict__ out, float* __restrict__ denom,
                         unsigned* __restrict__ maxbits, int nOut, int nDen) {
    int i = blockIdx.x * blockDim.x + threadIdx.x;
    int stride = gridDim.x * blockDim.x;
    for (int j = i; j < nOut; j += stride) out[j] = 0.0f;
    for (int j = i; j < nDen; j += stride) denom[j] = 0.0f;
    if (i == 0) *maxbits = 0u;   // encodes "below every real float"
}

// ------------------- K1: GEMM h = feat @ W via WMMA ------------------------
// One wave computes a 16-row x 64-col tile of h.
// W is staged TRANSPOSED in LDS: sWt[col*WT_STRIDE + k], so each B fragment
// (V0=K{k|k+2}, V1=K{k+1|k+3}, lane&15 = column) is one aligned ds_load_b64.
// A fragment (f32 16x4): lanes 0-15 hold rows M=lane, K={k,k+1};
//                        lanes 16-31 hold rows M=lane-16, K={k+2,k+3}.
__global__ __launch_bounds__(256)
void gat_gemm_wmma(const float* __restrict__ feat, const float* __restrict__ W,
                   float* __restrict__ h, int n) {
    __shared__ float sWt[GAT_HF * WT_STRIDE];   // 64*132*4 = 33792 B
    for (int i = threadIdx.x; i < GAT_INF * GAT_HF; i += 256) {
        int r = i >> 6;          // k row   0..127
        int c = i & 63;          // column  0..63
        sWt[c * WT_STRIDE + r] = W[i];
    }
    __syncthreads();

    const int wave = threadIdx.x >> 5;
    const int lane = threadIdx.x & 31;
    const int rowBase = (blockIdx.x * 8 + wave) * 16;
    if (rowBase >= n) return;                // wave-uniform: EXEC stays all-1s

    const int halfk = lane >> 4;             // 0 or 1
    const int l15   = lane & 15;
    int rowA = rowBase + l15;
    if (rowA >= n) rowA = n - 1;             // tail safety (n%16==0 normally)
    const float* aptr = feat + (size_t)rowA * GAT_INF + halfk * 2;

    // warm the near cache for this lane's feature row (global_prefetch_b8)
    __builtin_prefetch(aptr, 0, 0);
    __builtin_prefetch(aptr + 64, 0, 0);

    const float* bbase = sWt + (size_t)l15 * WT_STRIDE + halfk * 2;

    v8f acc0 = {}, acc1 = {}, acc2 = {}, acc3 = {};

    #pragma unroll
    for (int k = 0; k < GAT_INF; k += 4) {
        v2f a  = *(const v2f*)(aptr + k);                      // K = k+2h, k+2h+1
        v2f b0 = *(const v2f*)(bbase + k);                     // cols  0..15
        v2f b1 = *(const v2f*)(bbase + 16 * WT_STRIDE + k);    // cols 16..31
        v2f b2 = *(const v2f*)(bbase + 32 * WT_STRIDE + k);    // cols 32..47
        v2f b3 = *(const v2f*)(bbase + 48 * WT_STRIDE + k);    // cols 48..63

        acc0 = __builtin_amdgcn_wmma_f32_16x16x4_f32(false, a, false, b0,
                                                     (short)0, acc0, false, false);
        acc1 = __builtin_amdgcn_wmma_f32_16x16x4_f32(false, a, false, b1,
                                                     (short)0, acc1, false, false);
        acc2 = __builtin_amdgcn_wmma_f32_16x16x4_f32(false, a, false, b2,
                                                     (short)0, acc2, false, false);
        acc3 = __builtin_amdgcn_wmma_f32_16x16x4_f32(false, a, false, b3,
                                                     (short)0, acc3, false, false);
    }

    // C/D layout: VGPR r -> row rowBase + r + 8*halfk, col = l15 (+tile*16)
    #pragma unroll
    for (int r = 0; r < 8; ++r) {
        int row = rowBase + r + halfk * 8;
        if (row < n) {
            float* o = h + (size_t)row * GAT_HF + l15;
            o[0]  = acc0[r];
            o[16] = acc1[r];
            o[32] = acc2[r];
            o[48] = acc3[r];
        }
    }
}

// ------------------- K2: el / er projections -------------------------------
__global__ void gat_el_er(const float* __restrict__ h,
                          const float* __restrict__ attn_l,
                          const float* __restrict__ attn_r,
                          float* __restrict__ el, float* __restrict__ er, int n) {
    int i = blockIdx.x * blockDim.x + threadIdx.x;   // over n*HEADS
    if (i >= n * GAT_HEADS) return;
    int node = i >> 2, hd = i & 3;
    const float* hp = h + (size_t)node * GAT_HF + hd * GAT_OUTF;
    const float* al = attn_l + hd * GAT_OUTF;
    const float* ar = attn_r + hd * GAT_OUTF;
    float sl = 0.0f, sr = 0.0f;
    #pragma unroll
    for (int f = 0; f < GAT_OUTF; ++f) {
        float v = hp[f];
        sl += v * al[f];
        sr += v * ar[f];
    }
    el[i] = sl;
    er[i] = sr;
}

// ------------------- K3: global max of edge scores -------------------------
__global__ __launch_bounds__(256)
void gat_edge_max(const float* __restrict__ el, const float* __restrict__ er,
                  const int* __restrict__ src, const int* __restrict__ trg,
                  int E, unsigned* __restrict__ maxbits) {
    float m = -__builtin_inff();
    int stride = gridDim.x * blockDim.x;
    for (int e = blockIdx.x * blockDim.x + threadIdx.x; e < E; e += stride) {
        int s = src[e], t = trg[e];
        #pragma unroll
        for (int hd = 0; hd < GAT_HEADS; ++hd) {
            float v = el[s * GAT_HEADS + hd] + er[t * GAT_HEADS + hd];
            v = (v > 0.0f) ? v : GAT_NEG * v;
            m = fmaxf(m, v);
        }
    }
    #pragma unroll
    for (int off = 16; off > 0; off >>= 1)
        m = fmaxf(m, __shfl_xor(m, off, 32));
    __shared__ float sm[8];
    int lane = threadIdx.x & 31, wv = threadIdx.x >> 5;
    if (lane == 0) sm[wv] = m;
    __syncthreads();
    if (threadIdx.x == 0) {
        float bm = sm[0];
        #pragma unroll
        for (int i = 1; i < 8; ++i) bm = fmaxf(bm, sm[i]);
        atomicMax(maxbits, f32_to_ordered(bm));
    }
}

// ------------------- K4: e = exp(s - smax); denom += e ---------------------
__global__ void gat_edge_exp(const float* __restrict__ el, const float* __restrict__ er,
                             const int* __restrict__ src, const int* __restrict__ trg,
                             const unsigned* __restrict__ maxbits,
                             float* __restrict__ eArr, float* __restrict__ denom, int E) {
    int e = blockIdx.x * blockDim.x + threadIdx.x;
    if (e >= E) return;
    const float smax = ordered_to_f32(*maxbits);
    int s = src[e], t = trg[e];
    #pragma unroll
    for (int hd = 0; hd < GAT_HEADS; ++hd) {
        float v = el[s * GAT_HEADS + hd] + er[t * GAT_HEADS + hd];
        v = (v > 0.0f) ? v : GAT_NEG * v;
        float ex = __expf(v - smax);
        eArr[(size_t)e * GAT_HEADS + hd] = ex;
        __hip_atomic_fetch_add(&denom[t * GAT_HEADS + hd], ex,
                               __ATOMIC_RELAXED, __HIP_MEMORY_SCOPE_AGENT);
    }
}

// ------------------- K5: weighted scatter-aggregate ------------------------
// 16 lanes per edge; lane f handles feature f of all 4 heads.
__global__ void gat_aggregate(const float* __restrict__ h,
                              const float* __restrict__ eArr,
                              const float* __restrict__ denom,
                              const int* __restrict__ src, const int* __restrict__ trg,
                              float* __restrict__ out, int E) {
    unsigned tid = blockIdx.x * blockDim.x + threadIdx.x;
    int e = (int)(tid >> 4);
    int f = (int)(tid & 15u);
    if (e >= E) return;
    int s = src[e], t = trg[e];
    const float* hs = h + (size_t)s * GAT_HF;
    float* ot = out + (size_t)t * GAT_HF;
    #pragma unroll
    for (int hd = 0; hd < GAT_HEADS; ++hd) {
        float attn = eArr[(size_t)e * GAT_HEADS + hd] /
                     (denom[t * GAT_HEADS + hd] + GAT_EPS);
        float msg = hs[hd * GAT_OUTF + f] * attn;
        __hip_atomic_fetch_add(&ot[hd * GAT_OUTF + f], msg,
                               __ATOMIC_RELAXED, __HIP_MEMORY_SCOPE_AGENT);
    }
}

// ---------------------------------------------------------------------------
extern "C" void kernel_launch(void* const* d_in, const int* in_sizes, int n_in,
                              void* d_out, int out_size, void* d_ws, size_t ws_size,
                              hipStream_t stream) {
    const float* feat   = (const float*)d_in[0];
    const float* W      = (const float*)d_in[1];
    const float* attn_l = (const float*)d_in[2];
    const float* attn_r = (const float*)d_in[3];
    const int*   src    = (const int*)d_in[4];
    const int*   trg    = (const int*)d_in[5];
    float* out = (float*)d_out;

    const int n = in_sizes[0] / GAT_INF;   // N nodes
    const int E = in_sizes[4];             // edges

    // workspace layout (floats)
    float* ws    = (float*)d_ws;
    float* h     = ws;                                  // n*64
    float* el    = h     + (size_t)n * GAT_HF;          // n*4
    float* er    = el    + (size_t)n * GAT_HEADS;       // n*4
    float* denom = er    + (size_t)n * GAT_HEADS;       // n*4
    float* eArr  = denom + (size_t)n * GAT_HEADS;       // E*4
    unsigned* maxbits = (unsigned*)(eArr + (size_t)E * GAT_HEADS);

    gat_init<<<1024, 256, 0, stream>>>(out, denom, maxbits,
                                       n * GAT_HF, n * GAT_HEADS);

    const int waves = (n + 15) / 16;
    gat_gemm_wmma<<<(waves + 7) / 8, 256, 0, stream>>>(feat, W, h, n);

    gat_el_er<<<(n * GAT_HEADS + 255) / 256, 256, 0, stream>>>(h, attn_l, attn_r,
                                                               el, er, n);

    gat_edge_max<<<2048, 256, 0, stream>>>(el, er, src, trg, E, maxbits);

    gat_edge_exp<<<(E + 255) / 256, 256, 0, stream>>>(el, er, src, trg,
                                                      maxbits, eArr, denom, E);

    const long aggThreads = (long)E * 16;
    gat_aggregate<<<(int)((aggThreads + 255) / 256), 256, 0, stream>>>(
        h, eArr, denom, src, trg, out, E);
}